// GraphPool_13829794693589
// MI455X (gfx1250) — compile-verified
//
#include <hip/hip_runtime.h>

typedef float v2f __attribute__((ext_vector_type(2)));
typedef float v8f __attribute__((ext_vector_type(8)));

#define Bn 8
#define Nn 50000
#define Fn 256
#define Kn 25000
#define NB 65536
#define BN (Bn * Nn)

__device__ __forceinline__ unsigned orderKey(float f) {
    unsigned u = __float_as_uint(f);
    return (u & 0x80000000u) ? ~u : (u | 0x80000000u);
}

// ---- Kernel 1: scores = sigmoid(h . W + b) via V_WMMA_F32_16X16X4_F32 ----
// One wave per 16 nodes. A = 16x4 f32 tile of h (lane = M + 16*(K>>1), vgpr = K&1),
// B = W chunk replicated across all 16 columns (mirrored layout), so D[m][0] = dot.
__global__ void score_kernel(const float* __restrict__ h,
                             const float* __restrict__ W,
                             const float* __restrict__ bias,
                             float* __restrict__ scores,
                             unsigned* __restrict__ keys) {
    __shared__ float Wl[Fn];
    for (int i = threadIdx.x; i < Fn; i += blockDim.x) Wl[i] = W[i];
    __syncthreads();

    const int lane  = threadIdx.x & 31;
    const int tile  = blockIdx.x * (blockDim.x >> 5) + (threadIdx.x >> 5); // 25000 tiles exact
    const int m     = lane & 15;
    const int gn    = tile * 16 + m;          // flat node row (b*N+n)
    const int khalf = (lane >> 4) << 1;       // upper lane-half carries K=2,3

    const float* rowp = h + (size_t)gn * Fn;

    v8f c = {0.f, 0.f, 0.f, 0.f, 0.f, 0.f, 0.f, 0.f};
#pragma unroll 8
    for (int k0 = 0; k0 < Fn; k0 += 4) {
        int kk = k0 + khalf;
        v2f a  = *(const v2f*)(rowp + kk);    // A[m][kk..kk+1]
        v2f bv = *(const v2f*)(Wl + kk);      // B[kk..kk+1][*] = W (col 0 is what we use)
        c = __builtin_amdgcn_wmma_f32_16x16x4_f32(false, a, false, bv,
                                                  (short)0, c, false, false);
    }

    // D column N=0 lives in lane 0 (M=0..7 in c[0..7]) and lane 16 (M=8..15)
    if ((lane & 15) == 0) {
        float bb    = bias[0];
        int   mbase = (lane >> 4) * 8;
#pragma unroll
        for (int i = 0; i < 8; ++i) {
            int   node = tile * 16 + mbase + i;
            float x = c[i] + bb;
            float s = 1.0f / (1.0f + __expf(-x));
            scores[node] = s;
            keys[node]   = orderKey(s);
        }
    }
}

// ---- Kernel 0: zero the histogram/cursor buffer ----
__global__ void zero_kernel(unsigned* p, int n) {
    int i = blockIdx.x * blockDim.x + threadIdx.x;
    if (i < n) p[i] = 0u;
}

// ---- Kernel 2: 16-bit bucket histogram per batch ----
__global__ void hist_kernel(const unsigned* __restrict__ keys,
                            unsigned* __restrict__ hist) {
    int i = blockIdx.x * blockDim.x + threadIdx.x;
    if (i >= BN) return;
    int b = i / Nn;
    atomicAdd(&hist[b * NB + (keys[i] >> 16)], 1u);
}

// ---- Kernel 3: per-batch descending suffix scan; zero hist for cursor reuse ----
__global__ void scan_kernel(unsigned* __restrict__ hist,
                            unsigned* __restrict__ bstart) {
    __shared__ unsigned part[256];
    unsigned* hb = hist + blockIdx.x * NB;
    unsigned* sb = bstart + blockIdx.x * NB;
    int t = threadIdx.x;
    int base = t * 256;
    unsigned s = 0;
    for (int i = 0; i < 256; ++i) s += hb[base + i];
    part[t] = s;
    __syncthreads();
    if (t == 0) {
        unsigned run = 0;
        for (int j = 255; j >= 0; --j) { unsigned tmp = part[j]; part[j] = run; run += tmp; }
    }
    __syncthreads();
    unsigned run = part[t];                 // # keys in higher chunks
    for (int v = base + 255; v >= base; --v) {
        unsigned cnt = hb[v];
        sb[v] = run;                        // # keys with strictly higher bucket
        run += cnt;
        hb[v] = 0;                          // reset -> becomes scatter cursor
    }
}

// ---- Kernel 4: scatter node indices into per-bucket slot ranges ----
__global__ void scatter_kernel(const unsigned* __restrict__ keys,
                               const unsigned* __restrict__ bstart,
                               unsigned* __restrict__ cursor,
                               unsigned* __restrict__ slots) {
    int i = blockIdx.x * blockDim.x + threadIdx.x;
    if (i >= BN) return;
    int b = i / Nn;
    int n = i - b * Nn;
    unsigned v   = keys[i] >> 16;
    unsigned pos = atomicAdd(&cursor[b * NB + v], 1u);
    slots[(size_t)b * Nn + bstart[b * NB + v] + pos] = (unsigned)n;
}

// ---- Kernel 5: per-bucket exact sort (key desc, index asc) -> global rank ----
__global__ void sortrank_kernel(const unsigned* __restrict__ keys,
                                const unsigned* __restrict__ bstart,
                                const unsigned* __restrict__ cursor,
                                unsigned* __restrict__ slots,
                                unsigned* __restrict__ rank) {
    int i = blockIdx.x * blockDim.x + threadIdx.x;
    if (i >= Bn * NB) return;
    int b = i >> 16;
    unsigned cnt = cursor[i];
    if (cnt == 0) return;
    unsigned start = bstart[i];
    unsigned* sl = slots + (size_t)b * Nn + start;
    const unsigned* kb = keys + (size_t)b * Nn;
    for (unsigned j = 1; j < cnt; ++j) {       // tiny buckets: insertion sort
        unsigned nj = sl[j];
        unsigned kj = kb[nj];
        int p = (int)j - 1;
        while (p >= 0) {
            unsigned np = sl[p];
            unsigned kp = kb[np];
            if (kp > kj || (kp == kj && np < nj)) break;
            sl[p + 1] = np;
            --p;
        }
        sl[p + 1] = nj;
    }
    for (unsigned j = 0; j < cnt; ++j)
        rank[(size_t)b * Nn + sl[j]] = start + j;
}

// ---- Kernel 6: gather selected rows, scale by score, write in rank order ----
__global__ void gather_kernel(const float* __restrict__ h,
                              const float* __restrict__ scores,
                              const unsigned* __restrict__ rank,
                              float* __restrict__ out) {
    int row = blockIdx.x * 4 + (threadIdx.x >> 6);   // 64 threads per 256-f32 row
    if (row >= BN) return;
    unsigned r = rank[row];
    if (r >= Kn) return;                              // skip unselected: no h read
    int   b = row / Nn;
    float s = scores[row];
    const float4* src = (const float4*)(h + (size_t)row * Fn);
    float4*       dst = (float4*)(out + ((size_t)b * Kn + r) * Fn);
    int f = threadIdx.x & 63;
    float4 v = src[f];
    v.x *= s; v.y *= s; v.z *= s; v.w *= s;
    dst[f] = v;
}

extern "C" void kernel_launch(void* const* d_in, const int* in_sizes, int n_in,
                              void* d_out, int out_size, void* d_ws, size_t ws_size,
                              hipStream_t stream) {
    const float* h    = (const float*)d_in[0];
    const float* W    = (const float*)d_in[1];
    const float* bias = (const float*)d_in[2];
    float*       out  = (float*)d_out;

    float*    scores = (float*)d_ws;                  // BN f32
    unsigned* keys   = (unsigned*)(scores + BN);      // BN u32
    unsigned* hist   = keys + BN;                     // B*65536 (also cursor)
    unsigned* bstart = hist + Bn * NB;                // B*65536
    unsigned* slots  = bstart + Bn * NB;              // BN u32
    unsigned* rank   = slots + BN;                    // BN u32

    zero_kernel<<<(Bn * NB + 255) / 256, 256, 0, stream>>>(hist, Bn * NB);
    score_kernel<<<BN / 16 / 8, 256, 0, stream>>>(h, W, bias, scores, keys);
    hist_kernel<<<(BN + 255) / 256, 256, 0, stream>>>(keys, hist);
    scan_kernel<<<Bn, 256, 0, stream>>>(hist, bstart);
    scatter_kernel<<<(BN + 255) / 256, 256, 0, stream>>>(keys, bstart, hist, slots);
    sortrank_kernel<<<(Bn * NB + 255) / 256, 256, 0, stream>>>(keys, bstart, hist, slots, rank);
    gather_kernel<<<(BN + 3) / 4, 256, 0, stream>>>(h, scores, rank, out);
}